// EarlyExitGateLoss_14336600834433
// MI455X (gfx1250) — compile-verified
//
#include <hip/hip_runtime.h>
#include <hip/hip_bf16.h>
#include <stdint.h>

// ---------------------------------------------------------------------------
// Kernel 1: per-(b,k) row cross-entropy via log-softmax.
// One wave32 per row. The row (C=1000 floats, 4000 B) is streamed into LDS
// with CDNA5 async global->LDS copies (8 x b128 per wave, ASYNCcnt-tracked),
// then reduced in two passes (max, sum-exp) from LDS.
// ---------------------------------------------------------------------------

#define WAVES_PER_BLOCK 8
#define ROW_LDS_FLOATS  1024   // C <= 1024 (C == 1000 here), 4 KB per wave

__global__ __launch_bounds__(WAVES_PER_BLOCK * 32)
void eegl_ce_kernel(const float* __restrict__ yhats,
                    const int*   __restrict__ ys,
                    float*       __restrict__ ce_out,
                    int nRows, int C,
                    unsigned long long totalBytes)
{
    __shared__ float buf[WAVES_PER_BLOCK][ROW_LDS_FLOATS];

    const int wave = threadIdx.x >> 5;
    const int lane = threadIdx.x & 31;
    const int row  = blockIdx.x * WAVES_PER_BLOCK + wave;
    if (row >= nRows) return;

    // Wave-relative LDS byte offset of this wave's buffer: low 32 bits of the
    // generic (flat) address of an LDS object are the LDS byte address.
    const unsigned ldsBase =
        (unsigned)(unsigned long long)(uintptr_t)(&buf[wave][0]);

    const unsigned long long gBase   = (unsigned long long)(uintptr_t)yhats;
    const unsigned long long rowByte = (unsigned long long)row *
                                       (unsigned long long)C * 4ull;
    const unsigned long long clampHi = totalBytes - 16ull;

    const int rowBytes = C * 4;
    const int nChunks  = (rowBytes + 511) / 512;   // 8 for C=1000

    // Issue all async copies back-to-back: 8 in-flight 512B transfers/wave.
    for (int i = 0; i < nChunks; ++i) {
        const unsigned off = (unsigned)(i * 512 + lane * 16);
        unsigned long long gOff = rowByte + off;
        if (gOff > clampHi) gOff = clampHi;        // tail lanes: clamp (their
        const unsigned long long gAddr = gBase + gOff; // LDS slots are >= C
        const unsigned ldsAddr = ldsBase + off;        // and never read)
        asm volatile("global_load_async_to_lds_b128 %0, %1, off"
                     :: "v"(ldsAddr), "v"(gAddr)
                     : "memory");
    }
    // Wait for all async LDS writes of this wave.
    asm volatile("s_wait_asynccnt 0x0" ::: "memory");

    // Pass 1: row max.
    float m = -3.402823466e38f;
    for (int i = lane; i < C; i += 32) m = fmaxf(m, buf[wave][i]);
    #pragma unroll
    for (int o = 16; o > 0; o >>= 1) m = fmaxf(m, __shfl_xor(m, o, 32));

    // Pass 2: sum of exp(x - max).
    float s = 0.0f;
    for (int i = lane; i < C; i += 32) s += __expf(buf[wave][i] - m);
    #pragma unroll
    for (int o = 16; o > 0; o >>= 1) s += __shfl_xor(s, o, 32);

    if (lane == 0) {
        const int   tgt  = ys[row];                 // 0 <= tgt < C
        const float lse  = m + __logf(s);
        ce_out[row] = lse - buf[wave][tgt];         // -log_softmax[target]
    }
}

// ---------------------------------------------------------------------------
// Kernel 2: gate recurrence over K and final scalar reduction.
// Single 256-thread block; deterministic (no atomics); double accumulation.
//   gs_b   = sum_e p_reach*g*ce + p_last*ce[K-1]
//   cost_b = costs[first e with g>0.5] else costs[K-1]
//   out    = sum_b (0.5*gs_b + 0.5*cost_b)        (ALPHA = 0.5)
// ---------------------------------------------------------------------------

#define RED_THREADS 256

__global__ __launch_bounds__(RED_THREADS)
void eegl_gate_kernel(const float* __restrict__ ce,
                      const float* __restrict__ gconf,
                      const float* __restrict__ costs,
                      float*       __restrict__ out,
                      int B, int K)
{
    __shared__ double red[RED_THREADS];
    const int tid = threadIdx.x;
    const int E = K - 1;

    double acc = 0.0;
    for (int b = tid; b < B; b += RED_THREADS) {
        float p  = 1.0f;
        float gs = 0.0f;
        int firstIdx = -1;
        for (int e = 0; e < E; ++e) {
            const float ge = gconf[b * E + e];
            gs += p * ge * ce[b * K + e];
            p  *= (1.0f - ge);
            if (firstIdx < 0 && ge > 0.5f) firstIdx = e;
        }
        gs += p * ce[b * K + (K - 1)];                     // p_last * ce_last
        const float cost = (firstIdx >= 0) ? costs[firstIdx] : costs[K - 1];
        acc += 0.5 * (double)gs + 0.5 * (double)cost;      // ALPHA = 0.5
    }

    red[tid] = acc;
    __syncthreads();
    #pragma unroll
    for (int s = RED_THREADS / 2; s > 0; s >>= 1) {
        if (tid < s) red[tid] += red[tid + s];
        __syncthreads();
    }
    if (tid == 0) out[0] = (float)red[0];
}

// ---------------------------------------------------------------------------
// Launch
//   d_in[0] = ys               (int32,  B*K)
//   d_in[1] = y_hats           (float32, B*K*C)
//   d_in[2] = exit_confidences (float32, B*(K-1))
//   d_in[3] = costs            (float32, K)
//   d_out   = scalar float32
//   d_ws    = ce scratch (B*K floats = 192 KB)
// ---------------------------------------------------------------------------

extern "C" void kernel_launch(void* const* d_in, const int* in_sizes, int n_in,
                              void* d_out, int out_size, void* d_ws, size_t ws_size,
                              hipStream_t stream) {
    const int*   ys    = (const int*)d_in[0];
    const float* yhats = (const float*)d_in[1];
    const float* g     = (const float*)d_in[2];
    const float* costs = (const float*)d_in[3];
    float*       out   = (float*)d_out;
    float*       ce    = (float*)d_ws;

    const int K     = in_sizes[3];
    const int BK    = in_sizes[0];          // B*K rows
    const int B     = BK / K;
    const int C     = in_sizes[1] / BK;     // 1000
    const unsigned long long totalBytes = (unsigned long long)BK *
                                          (unsigned long long)C * 4ull;

    const int blocks = (BK + WAVES_PER_BLOCK - 1) / WAVES_PER_BLOCK;
    eegl_ce_kernel<<<blocks, WAVES_PER_BLOCK * 32, 0, stream>>>(
        yhats, ys, ce, BK, C, totalBytes);

    eegl_gate_kernel<<<1, RED_THREADS, 0, stream>>>(ce, g, costs, out, B, K);
}